// KCompetitiveLayer_53094385713529
// MI455X (gfx1250) — compile-verified
//
#include <hip/hip_runtime.h>

#define D      8192
#define NT     256
#define CHUNK  (D / NT)          // 32 elements per thread
#define KSEL   64u               // KTOP/2 per branch
#define ALPHA  6.26f

typedef unsigned int u32x4 __attribute__((ext_vector_type(4)));
typedef int          i32x4 __attribute__((ext_vector_type(4)));
typedef int          i32x8 __attribute__((ext_vector_type(8)));

// Pick the bin containing the k-th largest key at this radix level.
// Updates pref (accumulated high bits) and k (remaining rank within bin).
__device__ __forceinline__ void sel_level(const unsigned* hist, unsigned& pref, unsigned& k) {
  unsigned cum = 0;
  int b = 255;
  for (; b >= 0; --b) {
    cum += hist[b];
    if (cum >= k) break;
  }
  if (b < 0) b = 0;                 // safety (cannot happen: total >= k)
  k -= (cum - hist[b]);             // subtract count of strictly-greater bins
  pref = (pref << 8) | (unsigned)b;
}

__global__ __launch_bounds__(NT)
void kcomp_kernel(const float* __restrict__ x, float* __restrict__ out) {
  __shared__ __align__(16) float row[D];                       // 32 KB row tile
  __shared__ unsigned histP[256], histN[256];
  __shared__ unsigned cntP[256], cntN[256], prefP[256], prefN[256];
  __shared__ float red[256], red2[256];
  __shared__ unsigned sh_prefP, sh_prefN, sh_kP, sh_kN;
  __shared__ float sh_sumPos, sh_sumNeg, sh_gtSumIdx, sh_gtSumMag;
  __shared__ float sh_posTmp, sh_negTmp;

  const int t  = threadIdx.x;
  const int r0 = blockIdx.x;

  // ---------------- Stage 0: row -> LDS via Tensor Data Mover ----------------
#if defined(__gfx1250__) && __has_builtin(__builtin_amdgcn_tensor_load_to_lds)
  if (t < 32) {                      // wave 0 issues the DMA (EXEC ignored by TDM)
    unsigned long long ga = (unsigned long long)(x + (unsigned long long)r0 * D);
    // Generic LDS pointers carry the LDS byte offset in their low 32 bits.
    unsigned ldsOff = (unsigned)(unsigned long long)(const void*)&row[0];
    u32x4 g0;
    g0[0] = 1u;                                              // count=1, user mode
    g0[1] = ldsOff;                                          // lds_addr
    g0[2] = (unsigned)(ga & 0xFFFFFFFFull);                  // global_addr[31:0]
    g0[3] = ((unsigned)(ga >> 32) & 0x01FFFFFFu)             // global_addr[56:32]
            | 0x80000000u;                                   // type=2 ("image")
    i32x8 g1;
    g1[0] = 0x00020000;            // workgroup_mask=0, data_size=2 (4 bytes)
    g1[1] = (int)(8192u << 16);    // tensor_dim0[15:0]=8192 in bits[63:48]
    g1[2] = (int)(1u << 16);       // tensor_dim0 hi=0 ; tensor_dim1[15:0]=1
    g1[3] = (int)(8192u << 16);    // tensor_dim1 hi=0 ; tile_dim0=8192
    g1[4] = 1;                     // tile_dim1=1 ; tile_dim2=0
    g1[5] = 8192;                  // tensor_dim0_stride[31:0]=8192
    g1[6] = 0;                     // stride hi / tensor_dim1_stride lo
    g1[7] = 0;
    i32x4 g2 = {0, 0, 0, 0};
    i32x4 g3 = {0, 0, 0, 0};
#if __clang_major__ >= 23
    // amdgpu-toolchain / therock-10.0: 6-arg form (extra group operand, zeroed)
    i32x8 g4 = {0, 0, 0, 0, 0, 0, 0, 0};
    __builtin_amdgcn_tensor_load_to_lds(g0, g1, g2, g3, g4, 0);
#else
    // ROCm 7.2 clang-22: 5-arg form
    __builtin_amdgcn_tensor_load_to_lds(g0, g1, g2, g3, 0);
#endif
    __builtin_amdgcn_s_wait_tensorcnt(0);
  }
#else
  for (int j = 0; j < CHUNK; ++j) {
    int i = j * NT + t;
    row[i] = x[(unsigned long long)r0 * D + i];
  }
#endif
  histP[t] = 0;
  histN[t] = 0;
  __syncthreads();

  // ---------------- Pass A: level-0 histograms + branch sums -----------------
  {
    float sp = 0.f, sn = 0.f;
    for (int j = 0; j < CHUNK; ++j) {
      int i = j * NT + t;
      float v = row[i];
      unsigned bb = __float_as_uint(v);
      unsigned kp = (v > 0.f) ? bb : 0u;                  // key for relu(x)
      unsigned kn = (v < 0.f) ? (bb & 0x7FFFFFFFu) : 0u;  // key for |min(x,0)|
      if (v > 0.f) sp += v;
      if (v < 0.f) sn -= v;
      atomicAdd(&histP[kp >> 24], 1u);
      atomicAdd(&histN[kn >> 24], 1u);
    }
    red[t] = sp;
    red2[t] = sn;
  }
  __syncthreads();
  if (t == 0) {
    float a = 0.f, b2 = 0.f;
    for (int i = 0; i < NT; ++i) { a += red[i]; b2 += red2[i]; }
    sh_sumPos = a;
    sh_sumNeg = b2;
    sh_prefP = 0u; sh_kP = KSEL;
    sh_prefN = 0u; sh_kN = KSEL;
    sel_level(histP, sh_prefP, sh_kP);
    sel_level(histN, sh_prefN, sh_kN);
  }
  __syncthreads();

  // ---------------- Passes B: radix levels 1..3 ------------------------------
  for (int L = 1; L <= 3; ++L) {
    histP[t] = 0;
    histN[t] = 0;
    __syncthreads();
    const unsigned pP = sh_prefP, pN = sh_prefN;
    const int shM = 32 - 8 * L;   // compare shift
    const int shB = 24 - 8 * L;   // bin shift
    for (int j = 0; j < CHUNK; ++j) {
      int i = j * NT + t;
      float v = row[i];
      unsigned bb = __float_as_uint(v);
      unsigned kp = (v > 0.f) ? bb : 0u;
      unsigned kn = (v < 0.f) ? (bb & 0x7FFFFFFFu) : 0u;
      if ((kp >> shM) == pP) atomicAdd(&histP[(kp >> shB) & 0xFFu], 1u);
      if ((kn >> shM) == pN) atomicAdd(&histN[(kn >> shB) & 0xFFu], 1u);
    }
    __syncthreads();
    if (t == 0) {
      sel_level(histP, sh_prefP, sh_kP);
      sel_level(histN, sh_prefN, sh_kN);
    }
    __syncthreads();
  }

  const unsigned Tp = sh_prefP;       // exact 64th-largest pos key
  const unsigned Tn = sh_prefN;       // exact 64th-largest neg key
  const unsigned needP = sh_kP;       // # ties at Tp to take (lowest indices)
  const unsigned needN = sh_kN;

  // ---------------- Pass C1: tie counts per chunk + strict-> sums ------------
  {
    unsigned cp = 0, cn = 0;
    float gI = 0.f, gM = 0.f;
    for (int j = 0; j < CHUNK; ++j) {
      int i = t * CHUNK + j;          // index-contiguous chunk (ordered ties)
      float v = row[i];
      unsigned bb = __float_as_uint(v);
      unsigned kp = (v > 0.f) ? bb : 0u;
      unsigned kn = (v < 0.f) ? (bb & 0x7FFFFFFFu) : 0u;
      if (kp > Tp) gI += (float)i;        // bug path: scattered value = index
      else if (kp == Tp) ++cp;
      if (kn > Tn) gM += -v;              // magnitude
      else if (kn == Tn) ++cn;
    }
    cntP[t] = cp; cntN[t] = cn;
    red[t] = gI;  red2[t] = gM;
  }
  __syncthreads();
  if (t == 0) {
    unsigned aP = 0, aN = 0;
    float gI = 0.f, gM = 0.f;
    for (int i = 0; i < NT; ++i) {
      prefP[i] = aP; aP += cntP[i];
      prefN[i] = aN; aN += cntN[i];
      gI += red[i]; gM += red2[i];
    }
    sh_gtSumIdx = gI;
    sh_gtSumMag = gM;
  }
  __syncthreads();

  // ---------------- Pass C2: selected-tie index sum -> pos_tmp/neg_tmp -------
  {
    float tI = 0.f;
    unsigned r = prefP[t];
    if (r < needP && cntP[t] > 0) {
      for (int j = 0; j < CHUNK; ++j) {
        int i = t * CHUNK + j;
        float v = row[i];
        unsigned kp = (v > 0.f) ? __float_as_uint(v) : 0u;
        if (kp == Tp) {
          if (r < needP) tI += (float)i;
          ++r;
        }
      }
    }
    red[t] = tI;
  }
  __syncthreads();
  if (t == 0) {
    float tI = 0.f;
    for (int i = 0; i < NT; ++i) tI += red[i];
    float selIdx = sh_gtSumIdx + tI;
    sh_posTmp = ALPHA * (sh_sumPos - selIdx);
    float tieMag = __uint_as_float(Tn);          // all neg ties share |x|
    float selMag = sh_gtSumMag + (float)needN * tieMag;
    sh_negTmp = ALPHA * (sh_sumNeg - selMag);
  }
  __syncthreads();

  // ---------------- Pass C3: write output row into LDS -----------------------
  {
    const float posTmp = sh_posTmp, negTmp = sh_negTmp;
    const float tieMag = __uint_as_float(Tn);
    unsigned rP = prefP[t], rN = prefN[t];
    for (int j = 0; j < CHUNK; ++j) {
      int i = t * CHUNK + j;
      float v = row[i];
      unsigned bb = __float_as_uint(v);
      unsigned kp = (v > 0.f) ? bb : 0u;
      unsigned kn = (v < 0.f) ? (bb & 0x7FFFFFFFu) : 0u;
      float o = 0.f;
      if (kp > Tp)       o += (float)i + posTmp;
      else if (kp == Tp) { if (rP < needP) o += (float)i + posTmp; ++rP; }
      if (kn > Tn)       o -= (-v) + negTmp;
      else if (kn == Tn) { if (rN < needN) o -= tieMag + negTmp; ++rN; }
      row[i] = o;
    }
  }
  __syncthreads();

  // ---------------- Coalesced b128 store of the output row -------------------
  {
    const float4* src = (const float4*)row;
    float4* dst = (float4*)(out + (unsigned long long)r0 * D);
    for (int j = 0; j < D / 4 / NT; ++j) {
      int i = j * NT + t;
      dst[i] = src[i];
    }
  }
}

extern "C" void kernel_launch(void* const* d_in, const int* in_sizes, int n_in,
                              void* d_out, int out_size, void* d_ws, size_t ws_size,
                              hipStream_t stream) {
  (void)n_in; (void)out_size; (void)d_ws; (void)ws_size;
  const float* x = (const float*)d_in[0];
  float* out = (float*)d_out;
  int rows = in_sizes[0] / D;     // 8192 rows, one workgroup per row
  kcomp_kernel<<<dim3(rows), dim3(NT), 0, stream>>>(x, out);
}